// SetAbstraction_65910568124552
// MI455X (gfx1250) — compile-verified
//
#include <hip/hip_runtime.h>

typedef __attribute__((ext_vector_type(2))) float v2f;
typedef __attribute__((ext_vector_type(8))) float v8f;

namespace {

constexpr int BB = 2;
constexpr int NN = 16384;
constexpr int KK = 32;
constexpr int SSTR = 4;
constexpr int MM = NN / SSTR;    // 4096 centers per batch
constexpr int CIN = 64;
constexpr int CMID = 64;
constexpr int COUT = 128;
constexpr float R2 = 0.1f * 0.1f;

static __device__ __forceinline__ v8f wmma4(v2f a, v2f b, v8f c) {
  // D = A(16x4,f32) * B(4x16,f32) + C(16x16,f32)   -> v_wmma_f32_16x16x4_f32
  return __builtin_amdgcn_wmma_f32_16x16x4_f32(false, a, false, b, (short)0, c,
                                               false, false);
}

// Intra-wave LDS ordering: DS ops from one wave retire in order; force the
// split dscnt wait so cross-lane read-after-write through LDS is safe without
// a workgroup barrier (waves here own private LDS slices).
static __device__ __forceinline__ void lds_fence() {
  asm volatile("s_wait_dscnt 0" ::: "memory");
  __builtin_amdgcn_wave_barrier();
}

// Drain the gfx1250 async Global->LDS engine (ASYNCcnt) before DS reads.
static __device__ __forceinline__ void async_fence() {
  asm volatile("s_wait_asynccnt 0" ::: "memory");
  __builtin_amdgcn_wave_barrier();
}

// Async copy of 16 bytes/lane from global (SGPR base + 32-bit lane offset,
// GVS mode) directly into LDS at a per-lane byte address. ASYNCcnt-tracked.
static __device__ __forceinline__ void async_g2lds_b128(unsigned lds_byte_addr,
                                                        unsigned global_byte_off,
                                                        const float* base) {
  asm volatile("global_load_async_to_lds_b128 %0, %1, %2"
               :: "v"(lds_byte_addr), "v"(global_byte_off), "s"(base)
               : "memory");
}

static __device__ __forceinline__ void async_g2lds_b32(unsigned lds_byte_addr,
                                                       unsigned global_byte_off,
                                                       const float* base) {
  asm volatile("global_load_async_to_lds_b32 %0, %1, %2"
               :: "v"(lds_byte_addr), "v"(global_byte_off), "s"(base)
               : "memory");
}

} // namespace

// ---------------------------------------------------------------------------
// Kernel 1: ball query. One wave (32 lanes) per (b, m). Scans points in chunks
// of 32; ballot + prefix popcount appends the first K in-range indices in
// ascending index order (== reference top_k(-keymat) semantics). Pads with
// the first neighbor.
// ---------------------------------------------------------------------------
__global__ __launch_bounds__(256) void bq_kernel(const float* __restrict__ p,
                                                 int* __restrict__ nbr) {
  const int lane  = threadIdx.x & 31;
  const int wslot = threadIdx.x >> 5;
  const int wave  = blockIdx.x * 8 + wslot;          // BB*MM = 8192 waves
  const int b = wave / MM, m = wave % MM;
  const float* pb = p + (size_t)b * NN * 3;
  const int c = m * SSTR;
  const float cx = pb[c * 3 + 0], cy = pb[c * 3 + 1], cz = pb[c * 3 + 2];

  __shared__ int smem[8][KK];
  int* my = smem[wslot];

  int cnt = 0;
  for (int base = 0; base < NN && cnt < KK; base += 32) {
    const int idx = base + lane;
    const float dx = pb[idx * 3 + 0] - cx;
    const float dy = pb[idx * 3 + 1] - cy;
    const float dz = pb[idx * 3 + 2] - cz;
    const bool in = (dx * dx + dy * dy + dz * dz) <= R2;
#if __has_builtin(__builtin_amdgcn_ballot_w32)
    const unsigned mask = __builtin_amdgcn_ballot_w32(in);
#else
    const unsigned mask = (unsigned)__ballot(in);
#endif
    if (in) {
      const int pos = cnt + __popc(mask & ((1u << lane) - 1u));
      if (pos < KK) my[pos] = idx;
    }
    cnt += __popc(mask);                             // wave-uniform
  }
  lds_fence();
  if (cnt > KK) cnt = KK;
  int v;
  if (cnt == 0) v = c;                               // unreachable (self in ball)
  else          v = (lane < cnt) ? my[lane] : my[0];
  nbr[(size_t)wave * KK + lane] = v;
}

// ---------------------------------------------------------------------------
// Kernel 2: f1 = (w1_1 @ relu((w1_0 @ f)*s1_0 + t1_0))*s1_1 + t1_1
// One wave per 16-column tile of N. fp32 WMMA 16x16x4.
// B tile (64x16 of f) is staged ONCE into LDS via async Global->LDS B128
// copies (ASYNCcnt path), then reused by all 4 row tiles; h staged via LDS.
//   A layout: lane%16 = M-row, half selects K-pair, regs .x/.y = K within pair
//   C layout: lane%16 = N-col, reg j = row j (+8 for high half)
// ---------------------------------------------------------------------------
__global__ __launch_bounds__(256) void mlp1_kernel(
    const float* __restrict__ f,
    const float* __restrict__ w1_0, const float* __restrict__ s1_0,
    const float* __restrict__ t1_0,
    const float* __restrict__ w1_1, const float* __restrict__ s1_1,
    const float* __restrict__ t1_1,
    float* __restrict__ f1) {
  const int lane  = threadIdx.x & 31;
  const int wslot = threadIdx.x >> 5;
  const int l = lane & 15, hlf = lane >> 4;
  const int wave = blockIdx.x * 8 + wslot;           // 2048 waves
  const int tpb = NN / 16;                           // 1024 tiles per batch
  const int b  = wave / tpb;
  const int n0 = (wave % tpb) * 16;
  const float* fb = f + (size_t)b * CIN * NN;

  __shared__ float fbuf_all[8][CIN * 16];            // staged B tile
  __shared__ float hbuf_all[8][CMID * 16];           // staged h
  float* fbuf = fbuf_all[wslot];
  float* hbuf = hbuf_all[wslot];

  // Stage f[b][0:64][n0:n0+16] -> LDS with async B128 (16B/lane/op, 8 ops).
  {
    const unsigned lds_base = (unsigned)(uintptr_t)fbuf;
    for (int it = 0; it < 8; ++it) {
      const int g = it * 32 + lane;                  // dword4 id 0..255
      const int row = g >> 2, q = g & 3;             // row 0..63, 16B chunk
      const unsigned voff = (unsigned)((row * NN + n0) * 4 + q * 16);
      const unsigned ldsa = lds_base + (unsigned)(row * 64 + q * 16);
      async_g2lds_b128(ldsa, voff, fb);
    }
    async_fence();
  }

  for (int rt = 0; rt < 4; ++rt) {                   // h: 64 x 16
    v8f acc = {};
    const int r0 = rt * 16;
    for (int k = 0; k < CIN; k += 4) {
      const int ka = k + 2 * hlf;
      v2f a, bm;
      a.x  = w1_0[(r0 + l) * CIN + ka];
      a.y  = w1_0[(r0 + l) * CIN + ka + 1];
      bm.x = fbuf[ka * 16 + l];
      bm.y = fbuf[(ka + 1) * 16 + l];
      acc = wmma4(a, bm, acc);
    }
    for (int j = 0; j < 8; ++j) {
      const int row = r0 + j + 8 * hlf;
      hbuf[row * 16 + l] = fmaxf(acc[j] * s1_0[row] + t1_0[row], 0.0f);
    }
  }
  lds_fence();
  for (int rt = 0; rt < 8; ++rt) {                   // f1: 128 x 16
    v8f acc = {};
    const int r0 = rt * 16;
    for (int k = 0; k < CMID; k += 4) {
      const int ka = k + 2 * hlf;
      v2f a, bm;
      a.x  = w1_1[(r0 + l) * CMID + ka];
      a.y  = w1_1[(r0 + l) * CMID + ka + 1];
      bm.x = hbuf[ka * 16 + l];
      bm.y = hbuf[(ka + 1) * 16 + l];
      acc = wmma4(a, bm, acc);
    }
    for (int j = 0; j < 8; ++j) {
      const int row = r0 + j + 8 * hlf;
      f1[((size_t)b * COUT + row) * NN + n0 + l] =
          acc[j] * s1_1[row] + t1_1[row];            // no relu on f1
    }
  }
}

// ---------------------------------------------------------------------------
// Kernel 3: skip = w_skip @ f[:, :, centers] + b_skip  (128x64 GEMM, WMMA)
// Strided center columns staged into LDS via async B32 copies, reused by
// all 8 row tiles.
// ---------------------------------------------------------------------------
__global__ __launch_bounds__(256) void skip_kernel(
    const float* __restrict__ f, const float* __restrict__ w_skip,
    const float* __restrict__ b_skip, float* __restrict__ skipo) {
  const int lane  = threadIdx.x & 31;
  const int wslot = threadIdx.x >> 5;
  const int l = lane & 15, hlf = lane >> 4;
  const int wave = blockIdx.x * 8 + wslot;           // 512 waves
  const int tpb = MM / 16;                           // 256 tiles per batch
  const int b  = wave / tpb;
  const int m0 = (wave % tpb) * 16;
  const float* fb = f + (size_t)b * CIN * NN;

  __shared__ float fbuf_all[8][CIN * 16];
  float* fbuf = fbuf_all[wslot];

  // Stage f[b][0:64][stride-4 centers m0:m0+16] -> LDS (async b32 gather).
  {
    const unsigned lds_base = (unsigned)(uintptr_t)fbuf;
    for (int it = 0; it < 32; ++it) {
      const int g = it * 32 + lane;                  // 0..1023
      const int row = g >> 4, col = g & 15;
      const unsigned voff = (unsigned)((row * NN + (m0 + col) * SSTR) * 4);
      const unsigned ldsa = lds_base + (unsigned)((row * 16 + col) * 4);
      async_g2lds_b32(ldsa, voff, fb);
    }
    async_fence();
  }

  for (int rt = 0; rt < 8; ++rt) {
    v8f acc = {};
    const int r0 = rt * 16;
    for (int k = 0; k < CIN; k += 4) {
      const int ka = k + 2 * hlf;
      v2f a, bm;
      a.x  = w_skip[(r0 + l) * CIN + ka];
      a.y  = w_skip[(r0 + l) * CIN + ka + 1];
      bm.x = fbuf[ka * 16 + l];
      bm.y = fbuf[(ka + 1) * 16 + l];
      acc = wmma4(a, bm, acc);
    }
    for (int j = 0; j < 8; ++j) {
      const int row = r0 + j + 8 * hlf;
      skipo[((size_t)b * COUT + row) * MM + m0 + l] = acc[j] + b_skip[row];
    }
  }
}

// ---------------------------------------------------------------------------
// Kernel 4: fused positional MLP (3->32->32->128, WMMA for layers 2/3),
// f1 gather, max over K, +skip, relu. One wave per center, lane = neighbor.
// ---------------------------------------------------------------------------
__global__ __launch_bounds__(128) void pe_kernel(
    const float* __restrict__ p, const int* __restrict__ nbr,
    const float* __restrict__ f1, const float* __restrict__ skipo,
    const float* __restrict__ w2_0, const float* __restrict__ s2_0,
    const float* __restrict__ t2_0,
    const float* __restrict__ w2_1, const float* __restrict__ s2_1,
    const float* __restrict__ t2_1,
    const float* __restrict__ w2_2, const float* __restrict__ s2_2,
    const float* __restrict__ t2_2,
    float* __restrict__ out) {
  const int lane  = threadIdx.x & 31;
  const int wslot = threadIdx.x >> 5;
  const int l = lane & 15, hlf = lane >> 4;
  const int wave = blockIdx.x * 4 + wslot;           // 8192 waves
  const int b = wave / MM, m = wave % MM;

  __shared__ float x1T_all[4][32 * 32];              // [ch][pos]
  __shared__ float x2T_all[4][32 * 32];
  __shared__ int   sidx_all[4][KK];
  float* x1T = x1T_all[wslot];
  float* x2T = x2T_all[wslot];
  int*   sidx = sidx_all[wslot];

  const float* pb = p + (size_t)b * NN * 3;
  const int c = m * SSTR;
  const int idx = nbr[(size_t)wave * KK + lane];
  sidx[lane] = idx;
  const float dx = pb[idx * 3 + 0] - pb[c * 3 + 0];
  const float dy = pb[idx * 3 + 1] - pb[c * 3 + 1];
  const float dz = pb[idx * 3 + 2] - pb[c * 3 + 2];

  // layer 1: 3 -> 32 (VALU; weights scalarize to s-loads)
  for (int o = 0; o < 32; ++o) {
    const float v = w2_0[o * 3 + 0] * dx + w2_0[o * 3 + 1] * dy +
                    w2_0[o * 3 + 2] * dz;
    x1T[o * 32 + lane] = fmaxf(v * s2_0[o] + t2_0[o], 0.0f);
  }
  lds_fence();

  // layer 2: 32 -> 32 (WMMA, 32 ops)
  for (int rt = 0; rt < 2; ++rt) {
    for (int ct = 0; ct < 2; ++ct) {
      v8f acc = {};
      for (int k = 0; k < 32; k += 4) {
        const int ka = k + 2 * hlf;
        v2f a, bm;
        a.x  = w2_1[(rt * 16 + l) * 32 + ka];
        a.y  = w2_1[(rt * 16 + l) * 32 + ka + 1];
        bm.x = x1T[ka * 32 + ct * 16 + l];
        bm.y = x1T[(ka + 1) * 32 + ct * 16 + l];
        acc = wmma4(a, bm, acc);
      }
      for (int j = 0; j < 8; ++j) {
        const int row = rt * 16 + j + 8 * hlf;
        x2T[row * 32 + ct * 16 + l] =
            fmaxf(acc[j] * s2_1[row] + t2_1[row], 0.0f);
      }
    }
  }
  lds_fence();

  // layer 3: 32 -> 128 (WMMA, 128 ops) + f1 gather + running max over K
  for (int rt = 0; rt < 8; ++rt) {
    float rmax[8];
    for (int j = 0; j < 8; ++j) rmax[j] = -3.4e38f;
    for (int ct = 0; ct < 2; ++ct) {
      v8f acc = {};
      for (int k = 0; k < 32; k += 4) {
        const int ka = k + 2 * hlf;
        v2f a, bm;
        a.x  = w2_2[(rt * 16 + l) * 32 + ka];
        a.y  = w2_2[(rt * 16 + l) * 32 + ka + 1];
        bm.x = x2T[ka * 32 + ct * 16 + l];
        bm.y = x2T[(ka + 1) * 32 + ct * 16 + l];
        acc = wmma4(a, bm, acc);
      }
      const int gidx = sidx[ct * 16 + l];            // this column's neighbor
      for (int j = 0; j < 8; ++j) {
        const int row = rt * 16 + j + 8 * hlf;
        const float pe = fmaxf(acc[j] * s2_2[row] + t2_2[row], 0.0f);
        const float fj = f1[((size_t)b * COUT + row) * NN + gidx];
        rmax[j] = fmaxf(rmax[j], pe + fj);
      }
    }
    // max across the 16 positions held in each lane half (xor masks < 16)
    for (int off = 1; off < 16; off <<= 1)
      for (int j = 0; j < 8; ++j)
        rmax[j] = fmaxf(rmax[j], __shfl_xor(rmax[j], off, 32));
    if (l == 0) {                                    // lane 0 & lane 16 write
      for (int j = 0; j < 8; ++j) {
        const int row = rt * 16 + j + 8 * hlf;
        const float v = rmax[j] + skipo[((size_t)b * COUT + row) * MM + m];
        out[((size_t)b * COUT + row) * MM + m] = fmaxf(v, 0.0f);
      }
    }
  }
}

// ---------------------------------------------------------------------------
extern "C" void kernel_launch(void* const* d_in, const int* in_sizes, int n_in,
                              void* d_out, int out_size, void* d_ws,
                              size_t ws_size, hipStream_t stream) {
  (void)in_sizes; (void)n_in; (void)out_size; (void)ws_size;
  const float* p      = (const float*)d_in[0];
  const float* f      = (const float*)d_in[1];
  const float* w1_0   = (const float*)d_in[2];
  const float* s1_0   = (const float*)d_in[3];
  const float* t1_0   = (const float*)d_in[4];
  const float* w1_1   = (const float*)d_in[5];
  const float* s1_1   = (const float*)d_in[6];
  const float* t1_1   = (const float*)d_in[7];
  const float* w2_0   = (const float*)d_in[8];
  const float* s2_0   = (const float*)d_in[9];
  const float* t2_0   = (const float*)d_in[10];
  const float* w2_1   = (const float*)d_in[11];
  const float* s2_1   = (const float*)d_in[12];
  const float* t2_1   = (const float*)d_in[13];
  const float* w2_2   = (const float*)d_in[14];
  const float* s2_2   = (const float*)d_in[15];
  const float* t2_2   = (const float*)d_in[16];
  const float* w_skip = (const float*)d_in[17];
  const float* b_skip = (const float*)d_in[18];
  float* out = (float*)d_out;

  // workspace: nbr (1 MiB) | f1 (16 MiB) | skip (4 MiB)
  char* ws = (char*)d_ws;
  int*   nbr = (int*)ws;
  float* f1  = (float*)(ws + (1u << 20));
  float* skp = (float*)(ws + (1u << 20) + (16u << 20));

  bq_kernel<<<1024, 256, 0, stream>>>(p, nbr);
  mlp1_kernel<<<256, 256, 0, stream>>>(f, w1_0, s1_0, t1_0, w1_1, s1_1, t1_1,
                                       f1);
  skip_kernel<<<64, 256, 0, stream>>>(f, w_skip, b_skip, skp);
  pe_kernel<<<2048, 128, 0, stream>>>(p, nbr, f1, skp, w2_0, s2_0, t2_0, w2_1,
                                      s2_1, t2_1, w2_2, s2_2, t2_2, out);
}